// SelectiveSSM_8761733284417
// MI455X (gfx1250) — compile-verified
//
#include <hip/hip_runtime.h>
#include <math.h>

// Problem dims (fixed by the reference)
#define B_SZ     4
#define L_SEQ    4096
#define D_MODEL  1024
#define N_STATE  16
#define DT_RANK  64
#define M_ROWS   (B_SZ * L_SEQ)     // 16384
#define CHUNK    64
#define NCHUNK   (L_SEQ / CHUNK)    // 64

typedef __attribute__((ext_vector_type(2))) float v2f;
typedef __attribute__((ext_vector_type(8))) float v8f;

#if defined(__has_builtin)
#if __has_builtin(__builtin_amdgcn_wmma_f32_16x16x4_f32)
#define HAVE_WMMA_F32X4 1
#endif
#endif

#define USE_ASYNC_LDS 1

#if USE_ASYNC_LDS
// Per-lane 16B global->LDS async copy (gfx1250 GLOBAL_LOAD_ASYNC_TO_LDS_B128,
// tracked by ASYNCcnt). lds_byte_off is the LDS byte address for this lane.
__device__ __forceinline__ void async_copy_b128(unsigned lds_byte_off,
                                                const float* gaddr) {
    asm volatile("global_load_async_to_lds_b128 %0, %1, off"
                 :: "v"(lds_byte_off), "v"(gaddr) : "memory");
}
__device__ __forceinline__ void wait_async_all() {
    asm volatile("s_wait_asynccnt 0" ::: "memory");
}
__device__ __forceinline__ unsigned lds_off(const void* p) {
    return (unsigned)(unsigned long long)(__attribute__((address_space(3))) const char*)p;
}
#endif

// ---------------------------------------------------------------------------
// Kernel 0: u = silu(depthwise conv3 of x)   (B,L,D)
// ---------------------------------------------------------------------------
__global__ void k_conv_silu(const float* __restrict__ x,
                            const float* __restrict__ cw,
                            float* __restrict__ u) {
    int idx = blockIdx.x * 256 + threadIdx.x;        // over B*L*D
    int d   = idx % D_MODEL;
    int bl  = idx / D_MODEL;
    int l   = bl % L_SEQ;
    float xm = (l > 0)          ? x[idx - D_MODEL] : 0.0f;
    float xc = x[idx];
    float xp = (l < L_SEQ - 1)  ? x[idx + D_MODEL] : 0.0f;
    float v  = cw[d] * xm + cw[D_MODEL + d] * xc + cw[2 * D_MODEL + d] * xp;
    u[idx]   = v / (1.0f + __expf(-v));              // silu
}

// ---------------------------------------------------------------------------
// Kernel 1: [BC | dtr] = u @ [W_bc | W_dt] + bias
//   M = 16384, K = 1024, N = 96 (6 tiles of 16; tiles 0-1 -> BC, 2-5 -> dtr)
//   One wave per 16x16 output tile, v_wmma_f32_16x16x4_f32 along K.
// ---------------------------------------------------------------------------
__global__ void k_gemm_bcdt(const float* __restrict__ u,
                            const float* __restrict__ W_bc,
                            const float* __restrict__ b_bc,
                            const float* __restrict__ W_dt,
                            const float* __restrict__ b_dt,
                            float* __restrict__ BC,
                            float* __restrict__ dtr) {
    int wave = blockIdx.x * 8 + (threadIdx.x >> 5);
    int rt   = wave / 6;                 // row tile (0..1023)
    int ct   = wave % 6;                 // col tile (0..5), wave-uniform
    int lane = threadIdx.x & 31;
    int half = lane >> 4;
    int m    = lane & 15;
    int row  = rt * 16 + m;

    const float* W;  const float* bias;  int ncols, c;
    if (ct < 2) { W = W_bc; bias = b_bc; ncols = 32; c = ct * 16 + m; }
    else        { W = W_dt; bias = b_dt; ncols = 64; c = (ct - 2) * 16 + m; }

    v8f acc = {};
#ifdef HAVE_WMMA_F32X4
    int abase = row * D_MODEL + 2 * half;            // 32-bit offsets -> GVS addressing
    int bbase = 2 * half * ncols + c;
    for (int k0 = 0; k0 < D_MODEL; k0 += 4) {
        float2 t2 = *(const float2*)(u + abase + k0);
        v2f a;  a.x = t2.x;  a.y = t2.y;             // A[M=m][kb], A[M=m][kb+1]
        v2f bf; bf.x = W[bbase + k0 * ncols];        // B[kb][n=m]
                bf.y = W[bbase + (k0 + 1) * ncols];  // B[kb+1][n=m]
        acc = __builtin_amdgcn_wmma_f32_16x16x4_f32(false, a, false, bf,
                                                    (short)0, acc, false, false);
    }
#else
    for (int k0 = 0; k0 < D_MODEL; k0 += 4) {
        #pragma unroll
        for (int v = 0; v < 8; v++) {
            int ro = rt * 16 + v + half * 8;
            #pragma unroll
            for (int kk = 0; kk < 4; kk++)
                acc[v] += u[ro * D_MODEL + k0 + kk] * W[(k0 + kk) * ncols + c];
        }
    }
#endif
    float bs = bias[c];
    #pragma unroll
    for (int v = 0; v < 8; v++) {
        int rowo = rt * 16 + v + half * 8;           // D layout: VGPR v -> M = v (+8)
        float val = acc[v] + bs;
        if (ct < 2) BC [rowo * 32 + c] = val;
        else        dtr[rowo * 64 + c] = val;
    }
}

// ---------------------------------------------------------------------------
// Kernel 2: dt = softplus(dtr @ W_dtp + b_dtp)   M=16384, K=64, N=1024
// ---------------------------------------------------------------------------
__global__ void k_gemm_dt(const float* __restrict__ dtr,
                          const float* __restrict__ W_dtp,
                          const float* __restrict__ b_dtp,
                          float* __restrict__ dt) {
    int wave = blockIdx.x * 8 + (threadIdx.x >> 5);
    int rt   = wave >> 6;                // row tile (0..1023)
    int ct   = wave & 63;                // col tile (0..63)
    int lane = threadIdx.x & 31;
    int half = lane >> 4;
    int m    = lane & 15;
    int row  = rt * 16 + m;
    int col  = ct * 16 + m;

    v8f acc = {};
#ifdef HAVE_WMMA_F32X4
    int abase = row * DT_RANK + 2 * half;
    int bbase = 2 * half * D_MODEL + col;
    #pragma unroll
    for (int k0 = 0; k0 < DT_RANK; k0 += 4) {
        float2 t2 = *(const float2*)(dtr + abase + k0);
        v2f a;  a.x = t2.x;  a.y = t2.y;
        v2f bf; bf.x = W_dtp[bbase + k0 * D_MODEL];
                bf.y = W_dtp[bbase + (k0 + 1) * D_MODEL];
        acc = __builtin_amdgcn_wmma_f32_16x16x4_f32(false, a, false, bf,
                                                    (short)0, acc, false, false);
    }
#else
    for (int k0 = 0; k0 < DT_RANK; k0 += 4) {
        #pragma unroll
        for (int v = 0; v < 8; v++) {
            int ro = rt * 16 + v + half * 8;
            #pragma unroll
            for (int kk = 0; kk < 4; kk++)
                acc[v] += dtr[ro * DT_RANK + k0 + kk] * W_dtp[(k0 + kk) * D_MODEL + col];
        }
    }
#endif
    float bs = b_dtp[col];
    #pragma unroll
    for (int v = 0; v < 8; v++) {
        int rowo = rt * 16 + v + half * 8;
        float p  = acc[v] + bs;
        dt[rowo * D_MODEL + col] = (p > 20.0f) ? p : log1pf(__expf(p));  // softplus
    }
}

// ---------------------------------------------------------------------------
// Kernel 3 (pass A): per-chunk carries  G = exp(A * sum dt),  H = local scan end
//   block = (b, chunk, 256-wide d slab); B coeffs staged in LDS via async copy.
// ---------------------------------------------------------------------------
__global__ void k_chunk_gh(const float* __restrict__ x,
                           const float* __restrict__ dt,
                           const float* __restrict__ BC,
                           const float* __restrict__ A_log,
                           float* __restrict__ Gc,
                           float* __restrict__ Hc) {
    __shared__ float sB[CHUNK * N_STATE];                // 4 KB
    int blk  = blockIdx.x;
    int dblk = blk & 3;                                  // D/256 = 4 slabs
    int c    = (blk >> 2) & (NCHUNK - 1);
    int b    = blk >> 8;
    int tid  = threadIdx.x;
    int d    = dblk * 256 + tid;
    int rowbase = b * L_SEQ + c * CHUNK;

#if USE_ASYNC_LDS
    {   // 64 rows x 16 floats; each thread moves one 16B segment (256 total)
        int t   = tid >> 2;                              // chunk row 0..63
        int seg = tid & 3;                               // 4-float segment 0..3
        async_copy_b128(lds_off(&sB[t * 16 + seg * 4]),
                        BC + (rowbase + t) * 32 + seg * 4);
        wait_async_all();
    }
#else
    for (int i = tid; i < CHUNK * N_STATE; i += 256) {
        int t = i >> 4, n = i & 15;
        sB[i] = BC[(rowbase + t) * 32 + n];              // Bcoeff = BC[:, 0:16]
    }
#endif
    __syncthreads();

    float Ar[N_STATE], h[N_STATE];
    #pragma unroll
    for (int n = 0; n < N_STATE; n++) {
        Ar[n] = -__expf(A_log[d * N_STATE + n]);
        h[n]  = 0.0f;
    }
    float sdt = 0.0f;
    for (int t = 0; t < CHUNK; t++) {
        int ro = (rowbase + t) * D_MODEL + d;
        float dtv = dt[ro];
        float xd  = x[ro] * dtv;
        sdt += dtv;
        #pragma unroll
        for (int n = 0; n < N_STATE; n++) {
            float e = __expf(Ar[n] * dtv);
            h[n] = e * h[n] + sB[t * 16 + n] * xd;
        }
    }
    int ob = ((b * NCHUNK + c) * D_MODEL + d) * N_STATE;
    #pragma unroll
    for (int n = 0; n < N_STATE; n++) {
        Gc[ob + n] = __expf(Ar[n] * sdt);                // prod_t exp(A*dt_t)
        Hc[ob + n] = h[n];
    }
}

// ---------------------------------------------------------------------------
// Kernel 4 (pass B): sequential scan across chunk carries -> chunk start states
// ---------------------------------------------------------------------------
__global__ void k_chunk_scan(const float* __restrict__ Gc,
                             const float* __restrict__ Hc,
                             float* __restrict__ Hs) {
    int idx = blockIdx.x * 256 + threadIdx.x;            // over B*D*N = 65536
    int b   = idx / (D_MODEL * N_STATE);
    int dn  = idx % (D_MODEL * N_STATE);
    float h = 0.0f;
    for (int c = 0; c < NCHUNK; c++) {
        int o = (b * NCHUNK + c) * (D_MODEL * N_STATE) + dn;
        Hs[o] = h;                                       // state entering chunk c
        h = Gc[o] * h + Hc[o];
    }
}

// ---------------------------------------------------------------------------
// Kernel 5 (pass C): replay each chunk from its correct start state, emit y.
// ---------------------------------------------------------------------------
__global__ void k_chunk_out(const float* __restrict__ x,
                            const float* __restrict__ dt,
                            const float* __restrict__ BC,
                            const float* __restrict__ A_log,
                            const float* __restrict__ Dcoeff,
                            const float* __restrict__ Hs,
                            float* __restrict__ out) {
    __shared__ float sBC[CHUNK * 32];                    // 8 KB: B and C coeffs
    int blk  = blockIdx.x;
    int dblk = blk & 3;
    int c    = (blk >> 2) & (NCHUNK - 1);
    int b    = blk >> 8;
    int tid  = threadIdx.x;
    int d    = dblk * 256 + tid;
    int rowbase = b * L_SEQ + c * CHUNK;

#if USE_ASYNC_LDS
    {   // 2048 contiguous floats = 512 x 16B; 2 segments per thread
        #pragma unroll
        for (int k = 0; k < 2; k++) {
            int it = tid + k * 256;                      // 0..511
            async_copy_b128(lds_off(&sBC[it * 4]),
                            BC + rowbase * 32 + it * 4);
        }
        wait_async_all();
    }
#else
    for (int i = tid; i < CHUNK * 32; i += 256)
        sBC[i] = BC[rowbase * 32 + i];                   // contiguous chunk rows
#endif
    __syncthreads();

    float Ar[N_STATE], h[N_STATE];
    #pragma unroll
    for (int n = 0; n < N_STATE; n++)
        Ar[n] = -__expf(A_log[d * N_STATE + n]);
    int hb = ((b * NCHUNK + c) * D_MODEL + d) * N_STATE;
    #pragma unroll
    for (int n = 0; n < N_STATE; n++) h[n] = Hs[hb + n];
    float Dc = Dcoeff[d];

    for (int t = 0; t < CHUNK; t++) {
        int ro = (rowbase + t) * D_MODEL + d;
        float dtv = dt[ro];
        float xv  = x[ro];
        float xd  = xv * dtv;
        float y   = 0.0f;
        #pragma unroll
        for (int n = 0; n < N_STATE; n++) {
            float e = __expf(Ar[n] * dtv);
            h[n] = e * h[n] + sBC[t * 32 + n] * xd;      // B in cols 0..15
            y   += sBC[t * 32 + 16 + n] * h[n];          // C in cols 16..31
        }
        out[ro] = y + xv * Dc;
    }
}

// ---------------------------------------------------------------------------
extern "C" void kernel_launch(void* const* d_in, const int* in_sizes, int n_in,
                              void* d_out, int out_size, void* d_ws, size_t ws_size,
                              hipStream_t stream) {
    (void)in_sizes; (void)n_in; (void)out_size; (void)ws_size;
    const float* x     = (const float*)d_in[0];
    const float* convw = (const float*)d_in[1];
    const float* W_bc  = (const float*)d_in[2];
    const float* b_bc  = (const float*)d_in[3];
    const float* W_dt  = (const float*)d_in[4];
    const float* b_dt  = (const float*)d_in[5];
    const float* W_dtp = (const float*)d_in[6];
    const float* b_dtp = (const float*)d_in[7];
    const float* A_log = (const float*)d_in[8];
    const float* Dco   = (const float*)d_in[9];
    float* out = (float*)d_out;

    float* ws = (float*)d_ws;
    size_t o = 0;
    float* u   = ws + o; o += (size_t)M_ROWS * D_MODEL;                  // 67 MB
    float* dtb = ws + o; o += (size_t)M_ROWS * D_MODEL;                  // 67 MB
    float* dtr = ws + o; o += (size_t)M_ROWS * DT_RANK;                  //  4 MB
    float* BC  = ws + o; o += (size_t)M_ROWS * 32;                       //  2 MB
    float* Gc  = ws + o; o += (size_t)B_SZ * NCHUNK * D_MODEL * N_STATE; // 16 MB
    float* Hc  = ws + o; o += (size_t)B_SZ * NCHUNK * D_MODEL * N_STATE; // 16 MB
    float* Hs  = ws + o; o += (size_t)B_SZ * NCHUNK * D_MODEL * N_STATE; // 16 MB

    k_conv_silu<<<(M_ROWS * D_MODEL) / 256, 256, 0, stream>>>(x, convw, u);
    k_gemm_bcdt<<<(1024 * 6) / 8, 256, 0, stream>>>(u, W_bc, b_bc, W_dt, b_dt, BC, dtr);
    k_gemm_dt<<<(1024 * 64) / 8, 256, 0, stream>>>(dtr, W_dtp, b_dtp, dtb);
    k_chunk_gh<<<B_SZ * NCHUNK * 4, 256, 0, stream>>>(x, dtb, BC, A_log, Gc, Hc);
    k_chunk_scan<<<(B_SZ * D_MODEL * N_STATE) / 256, 256, 0, stream>>>(Gc, Hc, Hs);
    k_chunk_out<<<B_SZ * NCHUNK * 4, 256, 0, stream>>>(x, dtb, BC, A_log, Dco, Hs, out);
}